// DMPLayer_71021579206972
// MI455X (gfx1250) — compile-verified
//
#include <hip/hip_runtime.h>
#include <hip/hip_bf16.h>
#include <math.h>

typedef __attribute__((ext_vector_type(2))) float v2f;
typedef __attribute__((ext_vector_type(8))) float v8f;

#define TM  64    // rows per block tile
#define LDP 132   // LDS row stride (floats): banks 4m+k, half-waves disjoint mod 4

static __device__ __forceinline__ v8f wmma4(v2f a, v2f b, v8f c) {
  // D = A(16x4) * B(4x16) + C(16x16), fp32
  return __builtin_amdgcn_wmma_f32_16x16x4_f32(false, a, false, b, (short)0, c, false, false);
}

static __device__ __forceinline__ void atomAddG(float* p, float v) {
  __hip_atomic_fetch_add(p, v, __ATOMIC_RELAXED, __HIP_MEMORY_SCOPE_AGENT);
}
static __device__ __forceinline__ void atomAddL(float* p, float v) {
  __hip_atomic_fetch_add(p, v, __ATOMIC_RELAXED, __HIP_MEMORY_SCOPE_WORKGROUP);
}

// ---------------------------------------------------------------- init / degree / weight repack
__global__ void init_zero(float* p, long n) {
  long i = (long)blockIdx.x * blockDim.x + threadIdx.x;
  if (i < n) p[i] = 0.0f;
}

__global__ void degree_hist(const int* __restrict__ esrc, float* __restrict__ deg, long E) {
  long i = (long)blockIdx.x * blockDim.x + threadIdx.x;
  if (i < E) atomAddG(&deg[esrc[i]], 1.0f);
}

// Pack W (128x128 row-major) into pair-interleaved layout:
// P[p*128+n] = { W[2p][n], W[2p+1][n] },  p = 0..63.
// A WMMA B-fragment for k-step k (lane half hi, col n) is then the single
// 64-bit load P[((k>>1)+hi)*128 + n].
__global__ void pack_w(const float* __restrict__ W, v2f* __restrict__ P) {
  int i = blockIdx.x * blockDim.x + threadIdx.x;
  if (i < 64 * 128) {
    int p = i >> 7, n = i & 127;
    v2f v = { W[(2 * p) * 128 + n], W[(2 * p + 1) * 128 + n] };
    P[i] = v;
  }
}

// ---------------------------------------------------------------- edge pass 1
__global__ __launch_bounds__(128) void edge_pass1(
    const float* __restrict__ node_feat, const float* __restrict__ edge_feat,
    const int* __restrict__ edge_src, const int* __restrict__ edge_dst,
    const unsigned char* __restrict__ rev_flag,
    const v2f* __restrict__ pI,  const v2f* __restrict__ pO,
    const v2f* __restrict__ pS,  const v2f* __restrict__ pD,
    const v2f* __restrict__ pL,  const float* __restrict__ ebias,
    const v2f* __restrict__ pW1, const float* __restrict__ emlp_b1,
    const float* __restrict__ out_deg,
    float* __restrict__ agg, float* __restrict__ zE,
    float* __restrict__ esum, float* __restrict__ esumsq, long E)
{
  __shared__ float sT[TM][LDP];
  __shared__ int   sDst[TM];
  __shared__ int   sRev[TM];
  __shared__ float sFac[TM];
  __shared__ float sCS[128];
  __shared__ float sCQ[128];

  const int tid = threadIdx.x;
  const int wv  = tid >> 5;      // wave 0..3 -> 16-row strip
  const int lid = tid & 31;
  const int mi  = lid & 15;
  const int hi  = lid >> 4;
  const int kh  = hi * 2;
  const long e0 = (long)blockIdx.x * TM;

  if (tid < TM) {
    long e = e0 + tid; if (e >= E) e = E - 1;
    int d = edge_dst[e];
    sDst[tid] = d;
    sRev[tid] = rev_flag[e] ? 1 : 0;
    sFac[tid] = 2.0f * (1.0f + log2f(1.0f + out_deg[d]));
  }
  if (tid < 128) { sCS[tid] = 0.0f; sCQ[tid] = 0.0f; }
  __syncthreads();

  // per-lane A rows
  long e = e0 + wv * 16 + mi;
  long ec = (e < E) ? e : (E - 1);
  const int  es = edge_src[ec];
  const int  ed = edge_dst[ec];
  const bool rv = rev_flag[ec] != 0;
  const float* efr = edge_feat + ec * 128;
  const float* ar  = node_feat + (long)(rv ? es : ed) * 128;  // a = rev ? h_src : h_dst
  const float* br  = node_feat + (long)(rv ? ed : es) * 128;  // b = rev ? h_dst : h_src

  // hoist per-output-row metadata into registers (rows fixed per lane)
  long  dstOff[8];
  float facR[8];
  int   revR[8];
  int   valR[8];
#pragma unroll
  for (int r = 0; r < 8; ++r) {
    const int m = wv * 16 + r + 8 * hi;
    dstOff[r] = (long)sDst[m] * 128;
    facR[r]   = sFac[m];
    revR[r]   = sRev[m];
    valR[r]   = (e0 + m) < E;
  }

  for (int nt = 0; nt < 8; ++nt) {
    const int n = nt * 16 + mi;
    v8f cO = {}, cI = {}, cS = {}, cD = {}, cL = {}, cA = {}, cB = {};
    for (int k = 0; k < 128; k += 4) {
      const int kk = k + kh;
      const int bi = ((k >> 1) + hi) * 128 + n;
      v2f ae = *(const v2f*)(efr + kk);
      v2f aa = *(const v2f*)(ar  + kk);
      v2f ab = *(const v2f*)(br  + kk);
      v2f bO = pO[bi];
      v2f bI = pI[bi];
      v2f bS = pS[bi];
      v2f bD = pD[bi];
      v2f bL = pL[bi];
      cO = wmma4(ae, bO, cO);
      cI = wmma4(ae, bI, cI);
      cS = wmma4(ae, bS, cS);
      cD = wmma4(ae, bD, cD);
      cL = wmma4(ae, bL, cL);
      cA = wmma4(aa, bD, cA);   // a @ dst_w
      cB = wmma4(ab, bS, cB);   // b @ src_w
    }
    const float eb = ebias[n];
#pragma unroll
    for (int r = 0; r < 8; ++r) {
      const int m = wv * 16 + r + 8 * hi;
      if (valR[r]) {
        float msg = revR[r] ? cO[r] : -cI[r];
        atomAddG(&agg[dstOff[r] + n], msg);
      }
      sT[m][n] = cL[r] + facR[r] * (cS[r] - cD[r]) + (cA[r] - cB[r]) + eb;
    }
  }
  __syncthreads();

  // z = sT @ emlp_w1 + emlp_b1 ; store + BN stats
  for (int nt = 0; nt < 8; ++nt) {
    const int n = nt * 16 + mi;
    v8f cz = {};
    for (int k = 0; k < 128; k += 4) {
      const int kk = k + kh;
      v2f a = *(const v2f*)(&sT[wv * 16 + mi][kk]);
      v2f b = pW1[((k >> 1) + hi) * 128 + n];
      cz = wmma4(a, b, cz);
    }
    const float b1 = emlp_b1[n];
#pragma unroll
    for (int r = 0; r < 8; ++r) {
      const int m = wv * 16 + r + 8 * hi;
      if (valR[r]) {
        float z = cz[r] + b1;
        zE[(e0 + m) * 128 + n] = z;
        atomAddL(&sCS[n], z);
        atomAddL(&sCQ[n], z * z);
      }
    }
  }
  __syncthreads();
  if (tid < 128) {
    atomAddG(&esum[tid],   sCS[tid]);
    atomAddG(&esumsq[tid], sCQ[tid]);
  }
}

// ---------------------------------------------------------------- node pass 1
__global__ __launch_bounds__(128) void node_pass1(
    const float* __restrict__ node_feat, const float* __restrict__ agg,
    const v2f* __restrict__ pLoop, const float* __restrict__ nbias,
    const v2f* __restrict__ pW1, const float* __restrict__ nmlp_b1,
    float* __restrict__ zN, float* __restrict__ nsum, float* __restrict__ nsumsq, long N)
{
  __shared__ float sT[TM][LDP];
  __shared__ float sCS[128];
  __shared__ float sCQ[128];

  const int tid = threadIdx.x;
  const int wv  = tid >> 5;
  const int lid = tid & 31;
  const int mi  = lid & 15;
  const int hi  = lid >> 4;
  const int kh  = hi * 2;
  const long r0 = (long)blockIdx.x * TM;

  if (tid < 128) { sCS[tid] = 0.0f; sCQ[tid] = 0.0f; }
  __syncthreads();

  long row  = r0 + wv * 16 + mi;
  long rowc = (row < N) ? row : (N - 1);
  const float* nf = node_feat + rowc * 128;

  long aggOff[8];
  int  valR[8];
#pragma unroll
  for (int r = 0; r < 8; ++r) {
    const int m = wv * 16 + r + 8 * hi;
    long rr = r0 + m;
    valR[r]   = rr < N;
    aggOff[r] = ((rr < N) ? rr : (N - 1)) * 128;
  }

  for (int nt = 0; nt < 8; ++nt) {
    const int n = nt * 16 + mi;
    v8f c = {};
    for (int k = 0; k < 128; k += 4) {
      const int kk = k + kh;
      v2f a = *(const v2f*)(nf + kk);
      v2f b = pLoop[((k >> 1) + hi) * 128 + n];
      c = wmma4(a, b, c);
    }
    const float nb = nbias[n];
#pragma unroll
    for (int r = 0; r < 8; ++r) {
      const int m = wv * 16 + r + 8 * hi;
      sT[m][n] = c[r] + agg[aggOff[r] + n] + nb;
    }
  }
  __syncthreads();

  for (int nt = 0; nt < 8; ++nt) {
    const int n = nt * 16 + mi;
    v8f cz = {};
    for (int k = 0; k < 128; k += 4) {
      const int kk = k + kh;
      v2f a = *(const v2f*)(&sT[wv * 16 + mi][kk]);
      v2f b = pW1[((k >> 1) + hi) * 128 + n];
      cz = wmma4(a, b, cz);
    }
    const float b1 = nmlp_b1[n];
#pragma unroll
    for (int r = 0; r < 8; ++r) {
      const int m = wv * 16 + r + 8 * hi;
      if (valR[r]) {
        float z = cz[r] + b1;
        zN[(r0 + m) * 128 + n] = z;
        atomAddL(&sCS[n], z);
        atomAddL(&sCQ[n], z * z);
      }
    }
  }
  __syncthreads();
  if (tid < 128) {
    atomAddG(&nsum[tid],   sCS[tid]);
    atomAddG(&nsumsq[tid], sCQ[tid]);
  }
}

// ---------------------------------------------------------------- pass 2: BN+ReLU+GEMM (in place)
__global__ __launch_bounds__(128) void mlp_pass2(
    float* __restrict__ zbuf, const float* __restrict__ sum, const float* __restrict__ sumsq,
    float inv_cnt, const float* __restrict__ g, const float* __restrict__ be,
    const v2f* __restrict__ pW2, const float* __restrict__ b2, long R)
{
  __shared__ float sA[TM][LDP];
  __shared__ float sMu[128];
  __shared__ float sSc[128];

  const int tid = threadIdx.x;
  const int wv  = tid >> 5;
  const int lid = tid & 31;
  const int mi  = lid & 15;
  const int hi  = lid >> 4;
  const int kh  = hi * 2;
  const long r0 = (long)blockIdx.x * TM;

  if (tid < 128) {
    float mu  = sum[tid] * inv_cnt;
    float var = sumsq[tid] * inv_cnt - mu * mu;
    sMu[tid] = mu;
    sSc[tid] = rsqrtf(var + 1e-5f) * g[tid];
  }
  __syncthreads();

  const float muv = sMu[tid];
  const float scv = sSc[tid];
  const float bev = be[tid];
  for (int m = 0; m < TM; ++m) {
    long rr = r0 + m; long rc = (rr < R) ? rr : (R - 1);
    float v = zbuf[rc * 128 + tid];
    v = (v - muv) * scv + bev;
    sA[m][tid] = v > 0.0f ? v : 0.0f;
  }
  __syncthreads();

  int valR[8];
#pragma unroll
  for (int r = 0; r < 8; ++r) {
    const int m = wv * 16 + r + 8 * hi;
    valR[r] = (r0 + m) < R;
  }

  for (int nt = 0; nt < 8; ++nt) {
    const int n = nt * 16 + mi;
    v8f c = {};
    for (int k = 0; k < 128; k += 4) {
      const int kk = k + kh;
      v2f a = *(const v2f*)(&sA[wv * 16 + mi][kk]);
      v2f b = pW2[((k >> 1) + hi) * 128 + n];
      c = wmma4(a, b, c);
    }
    const float bb = b2[n];
#pragma unroll
    for (int r = 0; r < 8; ++r) {
      const int m = wv * 16 + r + 8 * hi;
      if (valR[r]) zbuf[(r0 + m) * 128 + n] = c[r] + bb;
    }
  }
}

// ---------------------------------------------------------------- launch
extern "C" void kernel_launch(void* const* d_in, const int* in_sizes, int n_in,
                              void* d_out, int out_size, void* d_ws, size_t ws_size,
                              hipStream_t stream) {
  const float* node_feat = (const float*)d_in[0];
  const float* edge_feat = (const float*)d_in[1];
  const int*   edge_src  = (const int*)d_in[2];
  const int*   edge_dst  = (const int*)d_in[3];
  const unsigned char* rev_flag = (const unsigned char*)d_in[4];  // jnp bool_ = 1 byte
  const float* in_w   = (const float*)d_in[5];
  const float* out_w  = (const float*)d_in[6];
  const float* src_w  = (const float*)d_in[7];
  const float* dst_w  = (const float*)d_in[8];
  const float* nloop_w= (const float*)d_in[9];
  const float* eloop_w= (const float*)d_in[10];
  const float* nbias  = (const float*)d_in[11];
  const float* ebias  = (const float*)d_in[12];
  const float* nmlp_w1= (const float*)d_in[13];
  const float* nmlp_b1= (const float*)d_in[14];
  const float* nmlp_g = (const float*)d_in[15];
  const float* nmlp_be= (const float*)d_in[16];
  const float* nmlp_w2= (const float*)d_in[17];
  const float* nmlp_b2= (const float*)d_in[18];
  const float* emlp_w1= (const float*)d_in[19];
  const float* emlp_b1= (const float*)d_in[20];
  const float* emlp_g = (const float*)d_in[21];
  const float* emlp_be= (const float*)d_in[22];
  const float* emlp_w2= (const float*)d_in[23];
  const float* emlp_b2= (const float*)d_in[24];

  const long N = in_sizes[0] / 128;
  const long E = in_sizes[1] / 128;

  float* ws     = (float*)d_ws;
  float* agg    = ws;                         // N*128
  float* deg    = agg + (size_t)N * 128;      // N
  float* esum   = deg + N;                    // 128
  float* esumsq = esum + 128;                 // 128
  float* nsum   = esumsq + 128;               // 128
  float* nsumsq = nsum + 128;                 // 128
  float* pack   = nsumsq + 128;               // 10 * 16384 floats (pair-interleaved weights)

  const float* wsrc[10] = { in_w, out_w, src_w, dst_w, eloop_w,
                            emlp_w1, nloop_w, nmlp_w1, emlp_w2, nmlp_w2 };
  v2f* pk[10];
  for (int i = 0; i < 10; ++i) pk[i] = (v2f*)(pack + (size_t)i * 16384);
  const v2f* pIn   = pk[0]; const v2f* pOut  = pk[1];
  const v2f* pSrc  = pk[2]; const v2f* pDst  = pk[3];
  const v2f* pEl   = pk[4]; const v2f* pEW1  = pk[5];
  const v2f* pNl   = pk[6]; const v2f* pNW1  = pk[7];
  const v2f* pEW2  = pk[8]; const v2f* pNW2  = pk[9];

  float* outF = (float*)d_out;
  float* zN = outF;                           // nout region (reused as z scratch)
  float* zE = outF + (size_t)N * 128;         // eout region (reused as z scratch)

  const long zeroCnt = (long)N * 128 + N + 512;
  init_zero<<<(unsigned)((zeroCnt + 255) / 256), 256, 0, stream>>>(ws, zeroCnt);
  for (int i = 0; i < 10; ++i)
    pack_w<<<(64 * 128 + 255) / 256, 256, 0, stream>>>(wsrc[i], pk[i]);
  degree_hist<<<(unsigned)((E + 255) / 256), 256, 0, stream>>>(edge_src, deg, E);

  edge_pass1<<<(unsigned)((E + TM - 1) / TM), 128, 0, stream>>>(
      node_feat, edge_feat, edge_src, edge_dst, rev_flag,
      pIn, pOut, pSrc, pDst, pEl, ebias,
      pEW1, emlp_b1, deg, agg, zE, esum, esumsq, E);

  node_pass1<<<(unsigned)((N + TM - 1) / TM), 128, 0, stream>>>(
      node_feat, agg, pNl, nbias, pNW1, nmlp_b1, zN, nsum, nsumsq, N);

  mlp_pass2<<<(unsigned)((E + TM - 1) / TM), 128, 0, stream>>>(
      zE, esum, esumsq, 1.0f / (float)E, emlp_g, emlp_be, pEW2, emlp_b2, E);

  mlp_pass2<<<(unsigned)((N + TM - 1) / TM), 128, 0, stream>>>(
      zN, nsum, nsumsq, 1.0f / (float)N, nmlp_g, nmlp_be, pNW2, nmlp_b2, N);
}